// MTGNN_53128745451590
// MI455X (gfx1250) — compile-verified
//
#include <hip/hip_runtime.h>
#include <math.h>

// ---------------------------------------------------------------------------
// MTGNN forward for MI455X (gfx1250, wave32).
// Heavy GEMMs (mixprop graph propagation + all convs) use V_WMMA_F32_16X16X4_F32
// with multi-accumulator tiles (8 WMMA per K-step in mixhop, 4 in convgemm).
// ---------------------------------------------------------------------------

typedef float v2f __attribute__((ext_vector_type(2)));
typedef float v8f __attribute__((ext_vector_type(8)));

#define TANHALPHA 3.0f
#define PROPALPHA 0.05f
#define LN_EPS    1e-5f

#define NN   512   // nodes
#define BB   64    // batch
#define RC   32    // residual/conv channels
#define SKIPC 64
#define ENDC 128
#define TT   64    // input time
#define NDIM 40
#define KTOP 20

#define WMMA_F32(a, b, c) \
  __builtin_amdgcn_wmma_f32_16x16x4_f32(false, (a), false, (b), (short)0, (c), false, false)

// ---------------------------------------------------------------------------
// Graph constructor: nv = tanh(alpha * (emb[idx] @ W^T + b))   [512 x 40]
// ---------------------------------------------------------------------------
__global__ void nv_kernel(const float* __restrict__ emb, const float* __restrict__ W,
                          const float* __restrict__ b, const int* __restrict__ idx,
                          float* __restrict__ nv) {
  int t = blockIdx.x * blockDim.x + threadIdx.x;
  if (t >= NN * NDIM) return;
  int i = t / NDIM, j = t - (t / NDIM) * NDIM;
  const float* e = emb + (size_t)idx[i] * NDIM;
  const float* w = W + (size_t)j * NDIM;
  float s = b[j];
  for (int d = 0; d < NDIM; ++d) s += e[d] * w[d];
  nv[t] = tanhf(TANHALPHA * s);
}

// adj = relu(tanh(alpha * (nv1 nv2^T - nv2 nv1^T)))   [512 x 512]
__global__ void adj_kernel(const float* __restrict__ nv1, const float* __restrict__ nv2,
                           float* __restrict__ adj) {
  int t = blockIdx.x * blockDim.x + threadIdx.x;
  if (t >= NN * NN) return;
  int i = t >> 9, j = t & (NN - 1);
  const float* a1 = nv1 + (size_t)i * NDIM;
  const float* b2 = nv2 + (size_t)j * NDIM;
  const float* a2 = nv2 + (size_t)i * NDIM;
  const float* b1 = nv1 + (size_t)j * NDIM;
  float s = 0.f;
  for (int d = 0; d < NDIM; ++d) s += a1[d] * b2[d] - a2[d] * b1[d];
  float v = tanhf(TANHALPHA * s);
  adj[t] = v > 0.f ? v : 0.f;
}

// Per-row top-K mask (one wave per row).
__global__ __launch_bounds__(32) void topk_kernel(float* __restrict__ adj) {
  __shared__ float row[NN];
  __shared__ float svals[32];
  __shared__ int   sidx[32];
  __shared__ int   keep[KTOP];
  int i = blockIdx.x;
  int lane = threadIdx.x;
  float* r = adj + (size_t)i * NN;
  for (int j = lane; j < NN; j += 32) row[j] = r[j];
  __syncthreads();
  for (int it = 0; it < KTOP; ++it) {
    float bv = -1e30f; int bi = 0;
    for (int j = lane; j < NN; j += 32) {
      float v = row[j];
      if (v > bv) { bv = v; bi = j; }
    }
    svals[lane] = bv; sidx[lane] = bi;
    __syncthreads();
    if (lane == 0) {
      float mb = -1e30f; int mi = 0;
      for (int l = 0; l < 32; ++l)
        if (svals[l] > mb) { mb = svals[l]; mi = sidx[l]; }
      keep[it] = mi;
      row[mi] = -1e30f;
    }
    __syncthreads();
  }
  for (int j = lane; j < NN; j += 32) {
    bool k = false;
    for (int it = 0; it < KTOP; ++it) if (keep[it] == j) k = true;
    if (!k) r[j] = 0.f;
  }
}

// Row-normalize (adj + I) and (adj^T + I).
__global__ void norm_adj_kernel(const float* __restrict__ adj,
                                float* __restrict__ aF, float* __restrict__ aB) {
  __shared__ float red[256];
  int v = blockIdx.x, t = threadIdx.x;
  float s = 0.f;
  for (int w = t; w < NN; w += 256) s += adj[(size_t)v * NN + w] + (w == v ? 1.f : 0.f);
  red[t] = s; __syncthreads();
  for (int o = 128; o > 0; o >>= 1) { if (t < o) red[t] += red[t + o]; __syncthreads(); }
  float rs = red[0]; __syncthreads();
  for (int w = t; w < NN; w += 256)
    aF[(size_t)v * NN + w] = (adj[(size_t)v * NN + w] + (w == v ? 1.f : 0.f)) / rs;
  float s2 = 0.f;
  for (int w = t; w < NN; w += 256) s2 += adj[(size_t)w * NN + v] + (w == v ? 1.f : 0.f);
  red[t] = s2; __syncthreads();
  for (int o = 128; o > 0; o >>= 1) { if (t < o) red[t] += red[t + o]; __syncthreads(); }
  float rs2 = red[0]; __syncthreads();
  for (int w = t; w < NN; w += 256)
    aB[(size_t)v * NN + w] = (adj[(size_t)w * NN + v] + (w == v ? 1.f : 0.f)) / rs2;
}

// ---------------------------------------------------------------------------
// Generic conv-as-GEMM with fp32 WMMA. One wave computes a 16x64 tile
// (1 m-tile x 4 column groups, 4 accumulators; A fragment reused 4x).
//   Y[b, coBase+co, n, to] = act( sum_{ci,kk} W[co,ci,kk] *
//                                 Xseg[b, ci%cps, n, shift+to+kk] + bias[co] )
// Columns p = n*To + to. K decomposition (ci,kk) tracked incrementally; the
// KW1 instantiation (1x1 convs, mixprop MLP) uses a pure linear row stride.
// ---------------------------------------------------------------------------
template <bool KW1>
__global__ __launch_bounds__(32) void convgemm_kernel(
    const float* __restrict__ X0, const float* __restrict__ X1,
    const float* __restrict__ X2, int cps,
    const float* __restrict__ W, const float* __restrict__ Bias,
    float* __restrict__ Y,
    int Cin, int Cout, int kw, int Ti, int To, int shift,
    int act, int accum, int coBase, int coStride)
{
  int b  = blockIdx.z;
  int m0 = blockIdx.y * 16;
  int p0 = blockIdx.x * 64;
  int P  = NN * To;
  int lane = threadIdx.x;
  int half = lane >> 4;
  int lm   = lane & 15;

  int cstride = NN * Ti;
  const float* Xb0 = X0 + (size_t)b * cps * cstride;
  const float* Xb1 = X1 + (size_t)b * cps * cstride;
  const float* Xb2 = X2 + (size_t)b * cps * cstride;

  // per-group column offsets (clamped for loads; stores guarded separately)
  int colOff[4];
#pragma unroll
  for (int g = 0; g < 4; ++g) {
    int p = p0 + g * 16 + lm; if (p >= P) p = P - 1;
    int n = p / To, to = p - n * To;
    colOff[g] = n * Ti + shift + to;
  }

  int co = m0 + lm; if (co >= Cout) co = Cout - 1;
  int Ktot = Cin * kw;
  const float* Wrow = W + (size_t)co * Ktot + 2 * half;

  // incremental (ci,kk) trackers for the two lane k-values
  int kA = 2 * half, kB = kA + 1;
  int cilA, kkA, cilB, kkB;
  if (KW1) { cilA = kA; kkA = 0; cilB = kB; kkB = 0; }
  else     { cilA = kA / kw; kkA = kA - cilA * kw;
             cilB = kB / kw; kkB = kB - cilB * kw; }
  int segA = 0, segB = 0;                 // cps >= 32 > 3, initial seg is 0
  int rowA = cilA * cstride + kkA;
  int rowB = cilB * cstride + kkB;

  v8f acc0 = {}, acc1 = {}, acc2 = {}, acc3 = {};
  for (int k0 = 0; k0 < Ktot; k0 += 4) {
    v2f afrag;
    afrag.x = Wrow[k0];
    afrag.y = Wrow[k0 + 1];
    const float* XpA = segA == 0 ? Xb0 : (segA == 1 ? Xb1 : Xb2);
    const float* XpB = segB == 0 ? Xb0 : (segB == 1 ? Xb1 : Xb2);
    v2f bf0, bf1, bf2, bf3;
    bf0.x = XpA[rowA + colOff[0]]; bf0.y = XpB[rowB + colOff[0]];
    bf1.x = XpA[rowA + colOff[1]]; bf1.y = XpB[rowB + colOff[1]];
    bf2.x = XpA[rowA + colOff[2]]; bf2.y = XpB[rowB + colOff[2]];
    bf3.x = XpA[rowA + colOff[3]]; bf3.y = XpB[rowB + colOff[3]];
    acc0 = WMMA_F32(afrag, bf0, acc0);
    acc1 = WMMA_F32(afrag, bf1, acc1);
    acc2 = WMMA_F32(afrag, bf2, acc2);
    acc3 = WMMA_F32(afrag, bf3, acc3);
    if (KW1) {
      cilA += 4; cilB += 4;
      rowA += 4 * cstride; rowB += 4 * cstride;
    } else {
      kkA += 4; kkB += 4;
#pragma unroll
      for (int u = 0; u < 4; ++u) {
        int wA = kkA >= kw; kkA = wA ? kkA - kw : kkA; cilA += wA;
        int wB = kkB >= kw; kkB = wB ? kkB - kw : kkB; cilB += wB;
      }
      rowA = cilA * cstride + kkA;
      rowB = cilB * cstride + kkB;
    }
    int sA = cilA >= cps;
    cilA = sA ? cilA - cps : cilA;
    rowA = sA ? rowA - cps * cstride : rowA;
    segA += sA;
    int sB = cilB >= cps;
    cilB = sB ? cilB - cps : cilB;
    rowB = sB ? rowB - cps * cstride : rowB;
    segB += sB;
  }

  // epilogue
  v8f accs[4] = {acc0, acc1, acc2, acc3};
#pragma unroll
  for (int g = 0; g < 4; ++g) {
    int pp = p0 + g * 16 + lm;
    if (pp >= P) continue;
    int n2 = pp / To, to2 = pp - n2 * To;
#pragma unroll
    for (int r = 0; r < 8; ++r) {
      int m = m0 + r + 8 * half;
      if (m < Cout) {
        float v = accs[g][r] + Bias[m];
        if (act == 1)      v = tanhf(v);
        else if (act == 2) v = 1.f / (1.f + __expf(-v));
        else if (act == 3) v = v > 0.f ? v : 0.f;
        size_t o = (((size_t)b * coStride + coBase + m) * NN + n2) * To + to2;
        if (accum) v += Y[o];
        Y[o] = v;
      }
    }
  }
}

// ---------------------------------------------------------------------------
// mixprop hop: Out[b,c,v,l] = alpha*X + (1-alpha)*sum_w A[v,w]*H[b,c,w,l]
// One wave computes a 32x64 tile (2 v-tiles x 4 l-tiles, 8 accumulators).
// Per K-step: running pointers (constant immediate offsets for the second
// v-tile / second K row), 10 loads, 8 WMMAs.
// ---------------------------------------------------------------------------
__global__ __launch_bounds__(32) void mixhop_kernel(
    const float* __restrict__ A, const float* __restrict__ H,
    const float* __restrict__ X, float* __restrict__ Out, int L)
{
  int bc = blockIdx.z;
  int v0 = blockIdx.y * 32;
  int l0 = blockIdx.x * 64;
  int lane = threadIdx.x;
  int half = lane >> 4, lm = lane & 15;

  const float* Hbc = H + (size_t)bc * NN * L;
  int lcol[4];
#pragma unroll
  for (int g = 0; g < 4; ++g) {
    int c = l0 + g * 16 + lm;
    lcol[g] = c < L ? c : L - 1;
  }
  // running pointers: A second v-tile at constant +16*NN (32KB immediate),
  // H second K-row at constant +L (<=256B immediate).
  const float* Ap  = A + (size_t)(v0 + lm) * NN + 2 * half;
  const float* Hp0 = Hbc + (size_t)(2 * half) * L + lcol[0];
  const float* Hp1 = Hbc + (size_t)(2 * half) * L + lcol[1];
  const float* Hp2 = Hbc + (size_t)(2 * half) * L + lcol[2];
  const float* Hp3 = Hbc + (size_t)(2 * half) * L + lcol[3];
  const size_t hstep = (size_t)4 * L;

  v8f a00 = {}, a01 = {}, a02 = {}, a03 = {};
  v8f a10 = {}, a11 = {}, a12 = {}, a13 = {};
#pragma unroll 2
  for (int k0 = 0; k0 < NN; k0 += 4) {
    v2f af0, af1, bf0, bf1, bf2, bf3;
    af0.x = Ap[0];       af0.y = Ap[1];
    af1.x = Ap[16 * NN]; af1.y = Ap[16 * NN + 1];
    bf0.x = Hp0[0]; bf0.y = Hp0[L];
    bf1.x = Hp1[0]; bf1.y = Hp1[L];
    bf2.x = Hp2[0]; bf2.y = Hp2[L];
    bf3.x = Hp3[0]; bf3.y = Hp3[L];
    a00 = WMMA_F32(af0, bf0, a00);
    a01 = WMMA_F32(af0, bf1, a01);
    a02 = WMMA_F32(af0, bf2, a02);
    a03 = WMMA_F32(af0, bf3, a03);
    a10 = WMMA_F32(af1, bf0, a10);
    a11 = WMMA_F32(af1, bf1, a11);
    a12 = WMMA_F32(af1, bf2, a12);
    a13 = WMMA_F32(af1, bf3, a13);
    Ap += 4;
    Hp0 += hstep; Hp1 += hstep; Hp2 += hstep; Hp3 += hstep;
  }

  const float* Xbc = X + (size_t)bc * NN * L;
  float* Obc = Out + (size_t)bc * NN * L;
  v8f accs[2][4] = {{a00, a01, a02, a03}, {a10, a11, a12, a13}};
#pragma unroll
  for (int tv = 0; tv < 2; ++tv) {
#pragma unroll
    for (int g = 0; g < 4; ++g) {
      int ll = l0 + g * 16 + lm;
      if (ll >= L) continue;
#pragma unroll
      for (int r = 0; r < 8; ++r) {
        int vv = v0 + tv * 16 + r + 8 * half;
        size_t o = (size_t)vv * L + ll;
        Obc[o] = PROPALPHA * Xbc[o] + (1.f - PROPALPHA) * accs[tv][g][r];
      }
    }
  }
}

// ---------------------------------------------------------------------------
// Elementwise kernels
// ---------------------------------------------------------------------------
__global__ void ew_mul_kernel(const float* __restrict__ a, const float* __restrict__ b,
                              float* __restrict__ o, size_t n) {
  size_t t = (size_t)blockIdx.x * blockDim.x + threadIdx.x;
  if (t < n) o[t] = a[t] * b[t];
}

__global__ void ew_relu_kernel(float* __restrict__ a, size_t n) {
  size_t t = (size_t)blockIdx.x * blockDim.x + threadIdx.x;
  if (t < n) { float v = a[t]; a[t] = v > 0.f ? v : 0.f; }
}

// x_new = m1 + m2 + x_old[..., -To:]
__global__ void combine_kernel(const float* __restrict__ m1, const float* __restrict__ m2,
                               const float* __restrict__ xold, float* __restrict__ o,
                               int Tprev, int To, size_t n) {
  size_t t = (size_t)blockIdx.x * blockDim.x + threadIdx.x;
  if (t >= n) return;
  size_t rest = t / (size_t)To;
  int to = (int)(t - rest * To);
  o[t] = m1[t] + m2[t] + xold[rest * Tprev + (Tprev - To) + to];
}

// ---------------------------------------------------------------------------
// Layer norm over (C,N,T) per batch element — deterministic 2-stage reduce.
// ---------------------------------------------------------------------------
__global__ void ln_reduce1_kernel(const float* __restrict__ x, float* __restrict__ part,
                                  int perB) {
  __shared__ float s1[256], s2[256];
  int b = blockIdx.y;
  const float* xb = x + (size_t)b * perB;
  float a = 0.f, q = 0.f;
  for (int t = blockIdx.x * 256 + threadIdx.x; t < perB; t += gridDim.x * 256) {
    float v = xb[t]; a += v; q += v * v;
  }
  s1[threadIdx.x] = a; s2[threadIdx.x] = q; __syncthreads();
  for (int o = 128; o > 0; o >>= 1) {
    if ((int)threadIdx.x < o) { s1[threadIdx.x] += s1[threadIdx.x + o]; s2[threadIdx.x] += s2[threadIdx.x + o]; }
    __syncthreads();
  }
  if (threadIdx.x == 0) {
    part[((size_t)b * 128 + blockIdx.x) * 2]     = s1[0];
    part[((size_t)b * 128 + blockIdx.x) * 2 + 1] = s2[0];
  }
}

__global__ void ln_reduce2_kernel(const float* __restrict__ part, float* __restrict__ stats) {
  __shared__ float s1[128], s2[128];
  int b = blockIdx.x, t = threadIdx.x;
  s1[t] = part[((size_t)b * 128 + t) * 2];
  s2[t] = part[((size_t)b * 128 + t) * 2 + 1];
  __syncthreads();
  for (int o = 64; o > 0; o >>= 1) {
    if (t < o) { s1[t] += s1[t + o]; s2[t] += s2[t + o]; }
    __syncthreads();
  }
  if (t == 0) { stats[b * 2] = s1[0]; stats[b * 2 + 1] = s2[0]; }
}

__global__ void ln_norm_kernel(const float* __restrict__ x, const float* __restrict__ stats,
                               const float* __restrict__ w, const float* __restrict__ bb,
                               const int* __restrict__ idx, float* __restrict__ y,
                               int To, float invCnt, size_t n) {
  size_t t = (size_t)blockIdx.x * blockDim.x + threadIdx.x;
  if (t >= n) return;
  size_t r = t / (size_t)To;  int to = (int)(t - r * To);
  size_t r2 = r / NN;         int nn = (int)(r - r2 * NN);
  size_t r3 = r2 / RC;        int c  = (int)(r2 - r3 * RC);
  int b = (int)r3;
  float mu  = stats[b * 2] * invCnt;
  float var = stats[b * 2 + 1] * invCnt - mu * mu;
  size_t wi = ((size_t)c * NN + idx[nn]) * To + to;
  y[t] = (x[t] - mu) * rsqrtf(var + LN_EPS) * w[wi] + bb[wi];
}

// ---------------------------------------------------------------------------
// Host-side orchestration
// ---------------------------------------------------------------------------
extern "C" void kernel_launch(void* const* d_in, const int* in_sizes, int n_in,
                              void* d_out, int out_size, void* d_ws, size_t ws_size,
                              hipStream_t stream) {
  (void)in_sizes; (void)n_in; (void)out_size; (void)ws_size;
  auto in = [&](int i) { return (const float*)d_in[i]; };

  // flat input order (insertion order of setup_inputs dicts):
  // 0:x 1:emb1 2:emb2 3:lin1_w 4:lin1_b 5:lin2_w 6:lin2_b
  // 7:start_w 8:start_b 9:skip0_w 10:skip0_b
  // layer j base=11+24*j: filt(w,b)x4, gate(w,b)x4, skip_w, skip_b,
  //                       g1.mlp_w, g1.mlp_b, g2.mlp_w, g2.mlp_b, norm_w, norm_b
  // 83:skipE_w 84:skipE_b 85:end1_w 86:end1_b 87:end2_w 88:end2_b 89:idx
  const float* x_in = in(0);
  const int* idx = (const int*)d_in[89];
  float* out = (float*)d_out;

  // workspace layout (floats)
  float* ws = (float*)d_ws;
  const size_t SZ = (size_t)BB * RC * NN * TT;
  float* bufX  = ws;
  float* bufF  = ws + SZ;
  float* bufG  = ws + 2 * SZ;
  float* bufH1 = ws + 3 * SZ;
  float* bufH2 = ws + 4 * SZ;
  float* bufM1 = ws + 5 * SZ;
  float* bufM2 = ws + 6 * SZ;
  float* skipAcc = ws + 7 * SZ;                      // [B,64,N,1]
  float* nv1   = skipAcc + (size_t)BB * SKIPC * NN;
  float* nv2   = nv1 + NN * NDIM;
  float* adj   = nv2 + NN * NDIM;
  float* aF    = adj + (size_t)NN * NN;
  float* aB    = aF + (size_t)NN * NN;
  float* lnPart = aB + (size_t)NN * NN;              // [B,128,2]
  float* lnStats = lnPart + BB * 128 * 2;            // [B,2]
  float* endBuf = lnStats + BB * 2;                  // [B,128,N,1]

  auto conv = [&](const float* X0, const float* X1, const float* X2, int cps,
                  const float* W, const float* Bi, float* Y,
                  int Cin, int Cout, int kw, int Ti, int To, int shift,
                  int act, int accum, int coBase, int coStride) {
    dim3 g((NN * To + 63) / 64, (Cout + 15) / 16, BB);
    if (kw == 1)
      convgemm_kernel<true><<<g, 32, 0, stream>>>(X0, X1, X2, cps, W, Bi, Y, Cin, Cout,
                                                  kw, Ti, To, shift, act, accum, coBase, coStride);
    else
      convgemm_kernel<false><<<g, 32, 0, stream>>>(X0, X1, X2, cps, W, Bi, Y, Cin, Cout,
                                                   kw, Ti, To, shift, act, accum, coBase, coStride);
  };
  auto mixhop = [&](const float* A, const float* H, const float* X, float* O, int L) {
    dim3 g((L + 63) / 64, NN / 32, BB * RC);
    mixhop_kernel<<<g, 32, 0, stream>>>(A, H, X, O, L);
  };

  // ---- graph constructor ----
  nv_kernel<<<(NN * NDIM + 255) / 256, 256, 0, stream>>>(in(1), in(3), in(4), idx, nv1);
  nv_kernel<<<(NN * NDIM + 255) / 256, 256, 0, stream>>>(in(2), in(5), in(6), idx, nv2);
  adj_kernel<<<(NN * NN + 255) / 256, 256, 0, stream>>>(nv1, nv2, adj);
  topk_kernel<<<NN, 32, 0, stream>>>(adj);
  norm_adj_kernel<<<NN, 256, 0, stream>>>(adj, aF, aB);

  // ---- start conv (1x1) + skip0 (full-T conv) ----
  conv(x_in, x_in, x_in, RC, in(7), in(8), bufX, RC, RC, 1, TT, TT, 0, 0, 0, 0, RC);
  conv(x_in, x_in, x_in, RC, in(9), in(10), skipAcc, RC, SKIPC, TT, TT, 1, 0, 0, 0, 0, SKIPC);

  // ---- layers ----
  const int kws[4] = {2, 3, 6, 7};
  int Tin = TT;
  for (int j = 0; j < 3; ++j) {
    int base = 11 + 24 * j;
    int To = Tin - 6;
    size_t nEl = (size_t)BB * RC * NN * To;
    // inception: filter (tanh) and gate (sigmoid), 4 branches of 8 channels
    for (int i = 0; i < 4; ++i)
      conv(bufX, bufX, bufX, RC, in(base + 2 * i), in(base + 2 * i + 1), bufF,
           RC, 8, kws[i], Tin, To, 7 - kws[i], 1, 0, 8 * i, RC);
    for (int i = 0; i < 4; ++i)
      conv(bufX, bufX, bufX, RC, in(base + 8 + 2 * i), in(base + 9 + 2 * i), bufG,
           RC, 8, kws[i], Tin, To, 7 - kws[i], 2, 0, 8 * i, RC);
    ew_mul_kernel<<<(unsigned)((nEl + 255) / 256), 256, 0, stream>>>(bufF, bufG, bufF, nEl);
    // skip conv (kernel width == To), accumulate into skipAcc
    conv(bufF, bufF, bufF, RC, in(base + 16), in(base + 17), skipAcc,
         RC, SKIPC, To, To, 1, 0, 0, 1, 0, SKIPC);
    // mixprop forward direction
    mixhop(aF, bufF, bufF, bufH1, To);
    mixhop(aF, bufH1, bufF, bufH2, To);
    conv(bufF, bufH1, bufH2, RC, in(base + 18), in(base + 19), bufM1,
         3 * RC, RC, 1, To, To, 0, 0, 0, 0, RC);
    // mixprop backward direction (A^T row-normalized)
    mixhop(aB, bufF, bufF, bufH1, To);
    mixhop(aB, bufH1, bufF, bufH2, To);
    conv(bufF, bufH1, bufH2, RC, in(base + 20), in(base + 21), bufM2,
         3 * RC, RC, 1, To, To, 0, 0, 0, 0, RC);
    // residual + layernorm
    combine_kernel<<<(unsigned)((nEl + 255) / 256), 256, 0, stream>>>(
        bufM1, bufM2, bufX, bufG, Tin, To, nEl);
    int perB = RC * NN * To;
    ln_reduce1_kernel<<<dim3(128, BB), 256, 0, stream>>>(bufG, lnPart, perB);
    ln_reduce2_kernel<<<BB, 128, 0, stream>>>(lnPart, lnStats);
    ln_norm_kernel<<<(unsigned)((nEl + 255) / 256), 256, 0, stream>>>(
        bufG, lnStats, in(base + 22), in(base + 23), idx, bufX, To, 1.f / perB, nEl);
    Tin = To;
  }

  // ---- output head ----
  conv(bufX, bufX, bufX, RC, in(83), in(84), skipAcc, RC, SKIPC, Tin, Tin, 1, 0, 0, 1, 0, SKIPC);
  size_t nS = (size_t)BB * SKIPC * NN;
  ew_relu_kernel<<<(unsigned)((nS + 255) / 256), 256, 0, stream>>>(skipAcc, nS);
  conv(skipAcc, skipAcc, skipAcc, SKIPC, in(85), in(86), endBuf,
       SKIPC, ENDC, 1, 1, 1, 0, 3, 0, 0, ENDC);
  conv(endBuf, endBuf, endBuf, ENDC, in(87), in(88), out,
       ENDC, 1, 1, 1, 1, 0, 0, 0, 0, 1);
}